// Query_model_13099650253436
// MI455X (gfx1250) — compile-verified
//
#include <hip/hip_runtime.h>
#include <math.h>
#include <stdint.h>

typedef __attribute__((ext_vector_type(2))) float v2f;
typedef __attribute__((ext_vector_type(8))) float v8f;

#define B_   64
#define T_   576
#define BT_  36864   // 64*576
#define C_   768
#define SD_  512
#define N_   8192
#define LN_EPS 1e-5f
#define INV_SQRT_SD 0.044194173824159216f  // 1/sqrt(512)

#define KROWS   12                 // w1 rows staged per async stage (3 K-steps)
#define NSTAGES (C_ / KROWS)       // 64
#define MT_PER_B (T_ / 16)         // 36 m-tiles per batch

__device__ __forceinline__ v8f wmma_f32(v2f a, v2f b, v8f c) {
  // V_WMMA_F32_16X16X4_F32 : D = A(16x4) * B(4x16) + C(16x16), exact fp32
  return __builtin_amdgcn_wmma_f32_16x16x4_f32(
      false, a, false, b, (short)0, c, false, false);
}

__device__ __forceinline__ float gelu_exact(float x) {
  return 0.5f * x * (1.0f + erff(x * 0.70710678118654752f));
}

// Stage one contiguous 24KB slab of w1 (12 rows x 512 cols) into LDS via the
// CDNA5 async global->LDS path (ASYNCcnt tracked). 96 bytes per thread.
__device__ __forceinline__ void stage_load_w1(const float* w1, float* dst_lds,
                                              int s, int tid) {
  unsigned long long ga =
      (unsigned long long)(uintptr_t)(w1 + (size_t)s * KROWS * SD_) +
      (unsigned long long)(tid * 96);
  unsigned la = (unsigned)(uintptr_t)dst_lds + (unsigned)(tid * 96);
  asm volatile(
      "global_load_async_to_lds_b128 %0, %1, off\n\t"
      "global_load_async_to_lds_b128 %0, %1, off offset:16\n\t"
      "global_load_async_to_lds_b128 %0, %1, off offset:32\n\t"
      "global_load_async_to_lds_b128 %0, %1, off offset:48\n\t"
      "global_load_async_to_lds_b128 %0, %1, off offset:64\n\t"
      "global_load_async_to_lds_b128 %0, %1, off offset:80"
      :: "v"(la), "v"(ga) : "memory");
}

__device__ __forceinline__ void wait_async0() {
  asm volatile("s_wait_asynccnt 0" ::: "memory");
}

// =====================================================================
// Fused kernel: LN1 -> GEMM1(w1)+b1 -> GELU -> LN2 -> masked token sum.
// One block (8 waves) owns a 16-row strip (one m-tile, single batch) and
// the full N=512 width; emits one 512-float partial of G. H never hits HBM.
// =====================================================================
__global__ void __launch_bounds__(256) fused1_kernel(
    const float* __restrict__ x, const int* __restrict__ mask,
    const float* __restrict__ g1, const float* __restrict__ bln1,
    const float* __restrict__ w1, const float* __restrict__ b1,
    float* __restrict__ Gpart) {
  __shared__ __align__(16) float slab[2][KROWS * SD_];   // 2 x 24KB
  __shared__ float mean1s[16], rstd1s[16], keepF[16];
  __shared__ float red_s[8][16], red_q[8][16];
  __shared__ float mean2s[16], rstd2s[16];

  const int tid  = threadIdx.x;
  const int wave = tid >> 5;
  const int lane = tid & 31;
  const int mt   = blockIdx.x;
  const int n0   = wave * 64;
  const int l16  = lane & 15;
  const int mhi  = lane >> 4;   // k-pair selector for A/B, m-half for C/D

  // --- kick stage 0 of the w1 slab pipeline immediately ---
  stage_load_w1(w1, &slab[0][0], 0, tid);

  // --- LN1 row stats for the 16 rows, cooperative (2 rows per wave) ---
  for (int rr = 0; rr < 2; rr++) {
    int r16 = wave * 2 + rr;
    const float* xr = x + (size_t)(mt * 16 + r16) * C_;
    float s = 0.f, q = 0.f;
    for (int i = lane; i < C_; i += 32) { float v = xr[i]; s += v; q += v * v; }
    for (int m = 16; m; m >>= 1) { s += __shfl_xor(s, m); q += __shfl_xor(q, m); }
    if (lane == 0) {
      float mm = s / (float)C_;
      mean1s[r16] = mm;
      rstd1s[r16] = rsqrtf(q / (float)C_ - mm * mm + LN_EPS);
    }
  }
  if (tid < 16) keepF[tid] = (mask[mt * 16 + tid] == 0) ? 1.f : 0.f;

  wait_async0();
  __syncthreads();   // stats + slab[0] ready

  const int row = mt * 16 + l16;
  const float mu = mean1s[l16];
  const float rs = rstd1s[l16];
  const float* xr = x + (size_t)row * C_;

  v8f c0 = {}, c1 = {}, c2 = {}, c3 = {};
  for (int s = 0; s < NSTAGES; s++) {
    int bi = s & 1;
    if (s + 1 < NSTAGES) stage_load_w1(w1, &slab[(s + 1) & 1][0], s + 1, tid);
#pragma unroll
    for (int kk = 0; kk < KROWS / 4; kk++) {
      int col = s * KROWS + kk * 4 + 2 * mhi;      // global k for this lane
      float2 xv = *(const float2*)(xr + col);
      float2 gv = *(const float2*)(g1 + col);
      float2 bv = *(const float2*)(bln1 + col);
      v2f a;
      a.x = (xv.x - mu) * rs * gv.x + bv.x;
      a.y = (xv.y - mu) * rs * gv.y + bv.y;
      const float* bp = &slab[bi][(kk * 4 + 2 * mhi) * SD_ + n0 + l16];
      v2f b;
      b.x = bp[0];  b.y = bp[SD_];       c0 = wmma_f32(a, b, c0);
      b.x = bp[16]; b.y = bp[SD_ + 16];  c1 = wmma_f32(a, b, c1);
      b.x = bp[32]; b.y = bp[SD_ + 32];  c2 = wmma_f32(a, b, c2);
      b.x = bp[48]; b.y = bp[SD_ + 48];  c3 = wmma_f32(a, b, c3);
    }
    wait_async0();
    __syncthreads();
  }

  // --- bias + exact GELU, in registers ---
  {
    float bb0 = b1[n0 + l16], bb1 = b1[n0 + 16 + l16];
    float bb2 = b1[n0 + 32 + l16], bb3 = b1[n0 + 48 + l16];
#pragma unroll
    for (int r = 0; r < 8; r++) {
      c0[r] = gelu_exact(c0[r] + bb0);
      c1[r] = gelu_exact(c1[r] + bb1);
      c2[r] = gelu_exact(c2[r] + bb2);
      c3[r] = gelu_exact(c3[r] + bb3);
    }
  }

  // --- LN2 row stats: per-lane partials, 16-lane shuffle reduce, LDS combine ---
  float p[8], pq[8];
#pragma unroll
  for (int r = 0; r < 8; r++) {
    float a0 = c0[r], a1 = c1[r], a2 = c2[r], a3 = c3[r];
    p[r]  = a0 + a1 + a2 + a3;
    pq[r] = a0 * a0 + a1 * a1 + a2 * a2 + a3 * a3;
  }
#pragma unroll
  for (int r = 0; r < 8; r++) {
    for (int m = 8; m; m >>= 1) {
      p[r]  += __shfl_xor(p[r], m);
      pq[r] += __shfl_xor(pq[r], m);
    }
  }
  if (l16 == 0) {
#pragma unroll
    for (int r = 0; r < 8; r++) {
      red_s[wave][r + 8 * mhi] = p[r];
      red_q[wave][r + 8 * mhi] = pq[r];
    }
  }
  __syncthreads();
  if (tid < 16) {
    float s = 0.f, q = 0.f;
    for (int w = 0; w < 8; w++) { s += red_s[w][tid]; q += red_q[w][tid]; }
    float mm = s / (float)SD_;
    mean2s[tid] = mm;
    rstd2s[tid] = rsqrtf(q / (float)SD_ - mm * mm + LN_EPS);
  }
  __syncthreads();

  // --- masked token-sum of LN2-normalized rows (g2/bln2 folded in later) ---
  float a0 = 0.f, a1 = 0.f, a2 = 0.f, a3 = 0.f;
#pragma unroll
  for (int r = 0; r < 8; r++) {
    int rr = r + 8 * mhi;
    float sc  = keepF[rr] * rstd2s[rr];
    float mu2 = mean2s[rr];
    a0 += sc * (c0[r] - mu2);
    a1 += sc * (c1[r] - mu2);
    a2 += sc * (c2[r] - mu2);
    a3 += sc * (c3[r] - mu2);
  }
  a0 += __shfl_xor(a0, 16);
  a1 += __shfl_xor(a1, 16);
  a2 += __shfl_xor(a2, 16);
  a3 += __shfl_xor(a3, 16);
  if (mhi == 0) {
    float* gp = Gpart + (size_t)mt * SD_ + n0 + l16;
    gp[0] = a0; gp[16] = a1; gp[32] = a2; gp[48] = a3;
  }
}

// ---------------- G[b,d] = g2[d]*sum_j Gpart + bln2[d]*cnt[b] ----------------
__global__ void finish_g_kernel(const float* __restrict__ Gpart,
                                const int* __restrict__ mask,
                                const float* __restrict__ g2, const float* __restrict__ bln2,
                                float* __restrict__ G, float* __restrict__ cnt) {
  int b = blockIdx.x;
  int d = threadIdx.x;  // blockDim = 512
  float s = 0.f;
  for (int j = 0; j < MT_PER_B; j++)
    s += Gpart[(size_t)(b * MT_PER_B + j) * SD_ + d];
  float c = 0.f;
  for (int t = 0; t < T_; t++) c += (mask[b * T_ + t] == 0) ? 1.f : 0.f;
  G[b * SD_ + d] = g2[d] * s + bln2[d] * c;
  if (d == 0) cnt[b] = c;
}

// ---------------- S = (G @ w2 + cnt*b2) / sqrt(512), [64,512]x[512,512] ----------------
__global__ void gemm2s_kernel(const float* __restrict__ G, const float* __restrict__ cnt,
                              const float* __restrict__ w2, const float* __restrict__ b2,
                              float* __restrict__ S) {
  const int NQ = SD_ / 64;
  int tid  = blockIdx.x * blockDim.x + threadIdx.x;
  int wid  = tid >> 5;
  int lane = tid & 31;
  int mt = wid / NQ;
  int n0 = (wid % NQ) * 64;
  int l16 = lane & 15, khalf = lane >> 4;
  int rowA = mt * 16 + l16;
  const float* gr = G + (size_t)rowA * SD_;

  v8f c0 = {}, c1 = {}, c2 = {}, c3 = {};
  for (int k0 = 0; k0 < SD_; k0 += 4) {
    int col = k0 + 2 * khalf;
    float2 av = *(const float2*)(gr + col);
    v2f a; a.x = av.x; a.y = av.y;
    const float* wp = w2 + (size_t)col * SD_ + n0 + l16;
    v2f b;
    b.x = wp[0];  b.y = wp[SD_];       c0 = wmma_f32(a, b, c0);
    b.x = wp[16]; b.y = wp[SD_ + 16];  c1 = wmma_f32(a, b, c1);
    b.x = wp[32]; b.y = wp[SD_ + 32];  c2 = wmma_f32(a, b, c2);
    b.x = wp[48]; b.y = wp[SD_ + 48];  c3 = wmma_f32(a, b, c3);
  }

  int mhi = lane >> 4;
  float bb0 = b2[n0 + l16], bb1 = b2[n0 + 16 + l16];
  float bb2 = b2[n0 + 32 + l16], bb3 = b2[n0 + 48 + l16];
#pragma unroll
  for (int r = 0; r < 8; r++) {
    int m = mt * 16 + r + 8 * mhi;
    float cn = cnt[m];
    float* o = S + (size_t)m * SD_ + n0 + l16;
    o[0]  = (c0[r] + cn * bb0) * INV_SQRT_SD;
    o[16] = (c1[r] + cn * bb1) * INV_SQRT_SD;
    o[32] = (c2[r] + cn * bb2) * INV_SQRT_SD;
    o[48] = (c3[r] + cn * bb3) * INV_SQRT_SD;
  }
}

// ---------------- pooled = S @ sd^T, [64,512] x [512,8192] ----------------
__global__ void gemm3_kernel(const float* __restrict__ S, const float* __restrict__ sd,
                             float* __restrict__ pooled) {
  const int NQ = N_ / 64;
  int tid  = blockIdx.x * blockDim.x + threadIdx.x;
  int wid  = tid >> 5;
  int lane = tid & 31;
  int mt = wid / NQ;
  int n0 = (wid % NQ) * 64;
  int l16 = lane & 15, khalf = lane >> 4;
  int rowA = mt * 16 + l16;
  const float* sr = S + (size_t)rowA * SD_;
  const float* sp0 = sd + (size_t)(n0 + l16) * SD_;
  const float* sp1 = sd + (size_t)(n0 + 16 + l16) * SD_;
  const float* sp2 = sd + (size_t)(n0 + 32 + l16) * SD_;
  const float* sp3 = sd + (size_t)(n0 + 48 + l16) * SD_;

  v8f c0 = {}, c1 = {}, c2 = {}, c3 = {};
  for (int k0 = 0; k0 < SD_; k0 += 4) {
    int col = k0 + 2 * khalf;
    float2 av = *(const float2*)(sr + col);
    v2f a; a.x = av.x; a.y = av.y;
    float2 bv; v2f b;
    bv = *(const float2*)(sp0 + col); b.x = bv.x; b.y = bv.y; c0 = wmma_f32(a, b, c0);
    bv = *(const float2*)(sp1 + col); b.x = bv.x; b.y = bv.y; c1 = wmma_f32(a, b, c1);
    bv = *(const float2*)(sp2 + col); b.x = bv.x; b.y = bv.y; c2 = wmma_f32(a, b, c2);
    bv = *(const float2*)(sp3 + col); b.x = bv.x; b.y = bv.y; c3 = wmma_f32(a, b, c3);
  }

  int mhi = lane >> 4;
#pragma unroll
  for (int r = 0; r < 8; r++) {
    int m = mt * 16 + r + 8 * mhi;
    float* o = pooled + (size_t)m * N_ + n0 + l16;
    o[0] = c0[r]; o[16] = c1[r]; o[32] = c2[r]; o[48] = c3[r];
  }
}

// ---------------- Softmax over N=8192 per row ----------------
__global__ void softmax_kernel(const float* __restrict__ pooled, float* __restrict__ outW) {
  int b = blockIdx.x;
  const float* p = pooled + (size_t)b * N_;
  __shared__ float sh[256];
  float mx = -3.402823466e38f;
  for (int i = threadIdx.x; i < N_; i += 256) mx = fmaxf(mx, p[i]);
  sh[threadIdx.x] = mx; __syncthreads();
  for (int off = 128; off > 0; off >>= 1) {
    if (threadIdx.x < off) sh[threadIdx.x] = fmaxf(sh[threadIdx.x], sh[threadIdx.x + off]);
    __syncthreads();
  }
  mx = sh[0]; __syncthreads();
  float s = 0.f;
  for (int i = threadIdx.x; i < N_; i += 256) s += expf(p[i] - mx);
  sh[threadIdx.x] = s; __syncthreads();
  for (int off = 128; off > 0; off >>= 1) {
    if (threadIdx.x < off) sh[threadIdx.x] += sh[threadIdx.x + off];
    __syncthreads();
  }
  float inv = 1.f / sh[0];
  for (int i = threadIdx.x; i < N_; i += 256)
    outW[(size_t)b * N_ + i] = expf(p[i] - mx) * inv;
}

// ---------------- att_ft = W @ sd, [64,8192] x [8192,512] ----------------
__global__ void gemm4_kernel(const float* __restrict__ W, const float* __restrict__ sd,
                             float* __restrict__ outFt) {
  const int NQ = SD_ / 64;
  int tid  = blockIdx.x * blockDim.x + threadIdx.x;
  int wid  = tid >> 5;
  int lane = tid & 31;
  int mt = wid / NQ;
  int n0 = (wid % NQ) * 64;
  int l16 = lane & 15, khalf = lane >> 4;
  int rowA = mt * 16 + l16;
  const float* wr = W + (size_t)rowA * N_;

  v8f c0 = {}, c1 = {}, c2 = {}, c3 = {};
  for (int k0 = 0; k0 < N_; k0 += 4) {
    int col = k0 + 2 * khalf;
    float2 av = *(const float2*)(wr + col);
    v2f a; a.x = av.x; a.y = av.y;
    const float* sp = sd + (size_t)col * SD_ + n0 + l16;
    v2f b;
    b.x = sp[0];  b.y = sp[SD_];       c0 = wmma_f32(a, b, c0);
    b.x = sp[16]; b.y = sp[SD_ + 16];  c1 = wmma_f32(a, b, c1);
    b.x = sp[32]; b.y = sp[SD_ + 32];  c2 = wmma_f32(a, b, c2);
    b.x = sp[48]; b.y = sp[SD_ + 48];  c3 = wmma_f32(a, b, c3);
  }

  int mhi = lane >> 4;
#pragma unroll
  for (int r = 0; r < 8; r++) {
    int m = mt * 16 + r + 8 * mhi;
    float* o = outFt + (size_t)m * SD_ + n0 + l16;
    o[0] = c0[r]; o[16] = c1[r]; o[32] = c2[r]; o[48] = c3[r];
  }
}

extern "C" void kernel_launch(void* const* d_in, const int* in_sizes, int n_in,
                              void* d_out, int out_size, void* d_ws, size_t ws_size,
                              hipStream_t stream) {
  const float* ft    = (const float*)d_in[0];
  const float* sd    = (const float*)d_in[1];
  const int*   mask  = (const int*)  d_in[2];
  const float* ln1_g = (const float*)d_in[3];
  const float* ln1_b = (const float*)d_in[4];
  const float* w1    = (const float*)d_in[5];
  const float* b1    = (const float*)d_in[6];
  const float* ln2_g = (const float*)d_in[7];
  const float* ln2_b = (const float*)d_in[8];
  const float* w2    = (const float*)d_in[9];
  const float* b2    = (const float*)d_in[10];

  float* outW  = (float*)d_out;                  // [64, 8192]
  float* outFt = outW + (size_t)B_ * N_;         // [64, 512]
  float* outSd = outFt + (size_t)B_ * SD_;       // [8192, 512]

  float* ws = (float*)d_ws;
  float* Gpart  = ws;                              // 2304 * 512
  float* G      = Gpart + (size_t)(BT_ / 16) * SD_;
  float* cnt    = G + (size_t)B_ * SD_;            // 64
  float* S      = cnt + B_;                        // 64*512
  float* pooled = S + (size_t)B_ * SD_;            // 64*8192

  // 1) fused LN1 -> GEMM1 -> GELU -> LN2 -> masked sum : one block per m-tile
  fused1_kernel<<<BT_ / 16, 256, 0, stream>>>(ft, mask, ln1_g, ln1_b, w1, b1, Gpart);

  // 2) combine partials -> G, cnt
  finish_g_kernel<<<B_, SD_, 0, stream>>>(Gpart, mask, ln2_g, ln2_b, G, cnt);

  // 3) S = (G @ w2 + cnt*b2) / sqrt(512)
  gemm2s_kernel<<<(B_ / 16) * (SD_ / 64) * 32 / 256, 256, 0, stream>>>(G, cnt, w2, b2, S);

  // 4) pooled = S @ sd^T
  gemm3_kernel<<<(B_ / 16) * (N_ / 64) * 32 / 256, 256, 0, stream>>>(S, sd, pooled);

  // 5) softmax -> att_weight (output 0)
  softmax_kernel<<<B_, 256, 0, stream>>>(pooled, outW);

  // 6) att_ft = att_weight @ sd (output 1)
  gemm4_kernel<<<(B_ / 16) * (SD_ / 64) * 32 / 256, 256, 0, stream>>>(outW, sd, outFt);

  // 7) sd passthrough (output 2)
  hipMemcpyAsync(outSd, sd, (size_t)N_ * SD_ * sizeof(float),
                 hipMemcpyDeviceToDevice, stream);
}